// RandomShift_61529701482797
// MI455X (gfx1250) — compile-verified
//
#include <hip/hip_runtime.h>
#include <hip/hip_bf16.h>

#define PADDING 4

typedef __attribute__((ext_vector_type(4))) float v4f;

// One block (8 wave32 waves) per (n,c) plane of 128x128 fp32.
// Wave w owns rows k*8+w, k=0..15. Each row is staged into a per-wave
// double-buffered LDS slot with global_load_async_to_lds_b128 (gfx1250
// async copy engine, tracked per-wave by ASYNCcnt -> no barriers needed),
// then column-shift-gathered with edge clamp and streamed out with
// 128-bit non-temporal stores.
__global__ __launch_bounds__(256) void RandomShift_61529701482797_kernel(
    const float* __restrict__ x, const int* __restrict__ shift,
    float* __restrict__ out, int C, int H)
{
    const int c     = blockIdx.x;          // channel
    const int n     = blockIdx.y;          // image  (no integer division)
    const int plane = n * C + c;
    const int tid   = threadIdx.x;
    const int wave  = tid >> 5;
    const int lane  = tid & 31;

    // integer pixel shift, folded with the replicate-pad offset
    const int sx = shift[2 * n + 0] - PADDING;   // column offset
    const int sy = shift[2 * n + 1] - PADDING;   // row offset

    const float* xp = x   + (size_t)plane * H * H;
    float*       op = out + (size_t)plane * H * H;

    __shared__ float smem[8][2][128];            // 8 waves x double buffer x row

    // LDS byte address of this wave's buffer pair (low 32 bits of flat addr)
    const unsigned lds_base = (unsigned)(size_t)(&smem[wave][0][0]);

    auto issue_row = [&](int k, int buf) {
        int i = k * 8 + wave;                    // output row
        int r = i + sy;                          // source row (edge clamp)
        r = r < 0 ? 0 : (r > H - 1 ? H - 1 : r);
        unsigned long long gaddr =
            (unsigned long long)(size_t)(xp + (size_t)r * H + lane * 4);
        unsigned laddr = lds_base + (unsigned)buf * 512u + (unsigned)lane * 16u;
        asm volatile("global_load_async_to_lds_b128 %0, %1, off th:TH_LOAD_NT"
                     :: "v"(laddr), "v"(gaddr)
                     : "memory");
    };

    // Loop-invariant clamped column indices (depend only on lane & sx).
    const int j0 = lane * 4;
    int c0 = j0 + 0 + sx; c0 = c0 < 0 ? 0 : (c0 > H - 1 ? H - 1 : c0);
    int c1 = j0 + 1 + sx; c1 = c1 < 0 ? 0 : (c1 > H - 1 ? H - 1 : c1);
    int c2 = j0 + 2 + sx; c2 = c2 < 0 ? 0 : (c2 > H - 1 ? H - 1 : c2);
    int c3 = j0 + 3 + sx; c3 = c3 < 0 ? 0 : (c3 > H - 1 ? H - 1 : c3);

    issue_row(0, 0);

    for (int k = 0; k < 16; ++k) {
        const int buf = k & 1;
        if (k + 1 < 16) {
            issue_row(k + 1, buf ^ 1);           // prefetch next row
            asm volatile("s_wait_asynccnt 1" ::: "memory");  // current row landed
        } else {
            asm volatile("s_wait_asynccnt 0" ::: "memory");
        }

        const float* row = &smem[wave][buf][0];
        v4f v;
        v.x = row[c0];
        v.y = row[c1];
        v.z = row[c2];
        v.w = row[c3];

        const int i = k * 8 + wave;
        __builtin_nontemporal_store(v, (v4f*)(op + (size_t)i * H + j0));
    }
}

extern "C" void kernel_launch(void* const* d_in, const int* in_sizes, int n_in,
                              void* d_out, int out_size, void* d_ws, size_t ws_size,
                              hipStream_t stream) {
    const float* x     = (const float*)d_in[0];
    const int*   shift = (const int*)d_in[1];
    float*       out   = (float*)d_out;

    const int H = 128;
    const int N = in_sizes[1] / 2;                 // 256
    const int C = in_sizes[0] / (N * H * H);       // 12

    dim3 grid(C, N);                               // one block per (n,c) plane
    dim3 block(256);                               // 8 wave32 waves
    hipLaunchKernelGGL(RandomShift_61529701482797_kernel, grid, block, 0, stream,
                       x, shift, out, C, H);
}